// CompressionAugmentation_86199993631147
// MI455X (gfx1250) — compile-verified
//
#include <hip/hip_runtime.h>
#include <stdint.h>
#include <math.h>

#define ROWLEN 4096
#define BLK 256

// ===================== Threefry2x32-20 (bit-exact JAX PRNG core) =====================
__host__ __device__ static inline void tf2x32(uint32_t k0, uint32_t k1,
                                              uint32_t c0, uint32_t c1,
                                              uint32_t* r0, uint32_t* r1) {
  uint32_t ks0 = k0, ks1 = k1, ks2 = 0x1BD11BDAu ^ k0 ^ k1;
  uint32_t x0 = c0 + ks0, x1 = c1 + ks1;
#define TF_R(rot) { x0 += x1; x1 = (x1 << (rot)) | (x1 >> (32 - (rot))); x1 ^= x0; }
#define TF_GA TF_R(13) TF_R(15) TF_R(26) TF_R(6)
#define TF_GB TF_R(17) TF_R(29) TF_R(16) TF_R(24)
  TF_GA x0 += ks1; x1 += ks2 + 1u;
  TF_GB x0 += ks2; x1 += ks0 + 2u;
  TF_GA x0 += ks0; x1 += ks1 + 3u;
  TF_GB x0 += ks1; x1 += ks2 + 4u;
  TF_GA x0 += ks2; x1 += ks0 + 5u;
#undef TF_R
#undef TF_GA
#undef TF_GB
  *r0 = x0; *r1 = x1;
}

// jax.random.split(key, 4): counters iota(8), halves (j, j+4)
static void jax_split4(uint32_t k0, uint32_t k1, uint32_t out8[8]) {
  for (int j = 0; j < 4; ++j) {
    uint32_t a, b;
    tf2x32(k0, k1, (uint32_t)j, (uint32_t)(j + 4), &a, &b);
    out8[j] = a;
    out8[4 + j] = b;
  }
}
// scalar random_bits: count iota(1) zero-padded -> block(0,0).v0
static uint32_t jax_bits_scalar(uint32_t k0, uint32_t k1) {
  uint32_t a, b; tf2x32(k0, k1, 0u, 0u, &a, &b); return a;
}
static float bits_to_u01_host(uint32_t bits) {
  uint32_t u = (bits >> 9) | 0x3f800000u;
  float f; __builtin_memcpy(&f, &u, 4);
  return f - 1.0f;
}

// ===================== db4 filters =====================
__constant__ float c_declo[8] = {
  -0.010597401785069032f, 0.0328830116668852f,   0.030841381835560764f, -0.18703481171909309f,
  -0.027983769416859854f, 0.6308807679298589f,   0.7148465705529157f,    0.2303778133088965f };

// ===================== WMMA f32 16x16x4 wrapper =====================
typedef float v2f __attribute__((ext_vector_type(2)));
typedef float v8f __attribute__((ext_vector_type(8)));

__device__ static inline v8f wmma4(v2f a, v2f b, v8f c) {
  // 8-arg form: (neg_a, A, neg_b, B, c_mod, C, reuse_a, reuse_b)
  return __builtin_amdgcn_wmma_f32_16x16x4_f32(false, a, false, b, (short)0, c, false, false);
}

// Per-lane fragments of the 16x16 DFT matrix F16[m][k] = exp(sgn*2i*pi*m*k/16),
// A-layout: row m = lane&15; K-halves split by lane>>4; 4 chunks of K=4.
struct Frag16 {
  v2f re[4];
  v2f im[4];
  v2f imn[4];
};

// ===================== radix-16 FFT (4096 = 16^3), DFT-16 via WMMA =====================
// Data laid out in LDS flat as idx = 256*d2 + 16*d1 + d0. Three stages of
// Y = F16 * M (WMMA, K accumulated in 4-chunks) + twiddles; stage 2 digit-reverses
// on store (deferred behind a barrier to avoid cross-wave RAW).
// Twiddles use native v_sin_f32/v_cos_f32 (TRANS pipe, co-executes with WMMA);
// |ang| < 6 rad so native range handling is fine, and ~1e-6 twiddle error is
// irrelevant for this augmentation pipeline.
template<int STAGE>
__device__ static inline void fft_stage(float* __restrict__ sre, float* __restrict__ sim,
                                        float sgn, const Frag16& F) {
  const int tid  = threadIdx.x;
  const int lane = tid & 31;
  const int wave = tid >> 5;          // 8 waves
  const int n    = lane & 15;         // B/D column
  const int half = lane >> 4;         // K-half / M-half selector

  if constexpr (STAGE < 2) {
#pragma unroll
    for (int tt = 0; tt < 2; ++tt) {
      const int t = wave + 8 * tt;    // tile 0..15 (16 columns each)
      v8f accRe = {0.f,0.f,0.f,0.f,0.f,0.f,0.f,0.f};
      v8f accIm = {0.f,0.f,0.f,0.f,0.f,0.f,0.f,0.f};
#pragma unroll
      for (int c = 0; c < 4; ++c) {
        const int k0 = 4 * c + 2 * half;
        int i0, step;
        if constexpr (STAGE == 0) { i0 = 256 * k0 + 16 * t + n; step = 256; }
        else                      { i0 = 256 * t + 16 * k0 + n; step = 16;  }
        v2f bRe, bIm;
        bRe.x = sre[i0]; bRe.y = sre[i0 + step];
        bIm.x = sim[i0]; bIm.y = sim[i0 + step];
        accRe = wmma4(F.re[c],  bRe, accRe);
        accRe = wmma4(F.imn[c], bIm, accRe);
        accIm = wmma4(F.re[c],  bIm, accIm);
        accIm = wmma4(F.im[c],  bRe, accIm);
      }
#pragma unroll
      for (int r = 0; r < 8; ++r) {
        const int u = r + 8 * half;   // output digit (D row)
        float dRe = accRe[r], dIm = accIm[r];
        float ang; int idx;
        if constexpr (STAGE == 0) {   // twiddle W256^(b*u), b = t
          ang = sgn * 6.2831853071795864f * (float)(t * u) * (1.0f / 256.0f);
          idx = 256 * u + 16 * t + n;
        } else {                      // twiddle W4096^(c*(u0+16*v)), c=n, u0=t, v=u
          ang = sgn * 6.2831853071795864f * (float)(n * (t + 16 * u)) * (1.0f / 4096.0f);
          idx = 256 * t + 16 * u + n;
        }
        float cw = __cosf(ang), sw = __sinf(ang);   // native TRANS ops
        sre[idx] = dRe * cw - dIm * sw;
        sim[idx] = dRe * sw + dIm * cw;
      }
    }
  } else {
    v8f keepRe[2], keepIm[2];
#pragma unroll
    for (int tt = 0; tt < 2; ++tt) {
      const int t = wave + 8 * tt;
      v8f accRe = {0.f,0.f,0.f,0.f,0.f,0.f,0.f,0.f};
      v8f accIm = {0.f,0.f,0.f,0.f,0.f,0.f,0.f,0.f};
#pragma unroll
      for (int c = 0; c < 4; ++c) {
        const int k0 = 4 * c + 2 * half;
        const int i0 = 256 * t + 16 * n + k0;
        v2f bRe, bIm;
        bRe.x = sre[i0]; bRe.y = sre[i0 + 1];
        bIm.x = sim[i0]; bIm.y = sim[i0 + 1];
        accRe = wmma4(F.re[c],  bRe, accRe);
        accRe = wmma4(F.imn[c], bIm, accRe);
        accIm = wmma4(F.re[c],  bIm, accIm);
        accIm = wmma4(F.im[c],  bRe, accIm);
      }
      keepRe[tt] = accRe; keepIm[tt] = accIm;
    }
    __syncthreads();                  // all reads done before digit-reversed scatter
#pragma unroll
    for (int tt = 0; tt < 2; ++tt) {
      const int t = wave + 8 * tt;
#pragma unroll
      for (int r = 0; r < 8; ++r) {
        const int w = r + 8 * half;
        const int idx = t + 16 * n + 256 * w;   // X[u + 16v + 256w], u=t, v=n
        sre[idx] = keepRe[tt][r];
        sim[idx] = keepIm[tt][r];
      }
    }
  }
  __syncthreads();
}

__device__ static inline void fft4096(float* __restrict__ sre, float* __restrict__ sim,
                                      float sgn) {
  const int lane = threadIdx.x & 31;
  const int n    = lane & 15;
  const int half = lane >> 4;
  Frag16 F;
#pragma unroll
  for (int c = 0; c < 4; ++c) {
    const int k0 = 4 * c + 2 * half;
    float a0 = sgn * 0.39269908169872414f * (float)((n * k0) & 15);
    float a1 = sgn * 0.39269908169872414f * (float)((n * (k0 + 1)) & 15);
    F.re[c].x  = __cosf(a0); F.re[c].y  = __cosf(a1);
    F.im[c].x  = __sinf(a0); F.im[c].y  = __sinf(a1);
    F.imn[c].x = -F.im[c].x; F.imn[c].y = -F.im[c].y;
  }
  fft_stage<0>(sre, sim, sgn, F);
  fft_stage<1>(sre, sim, sgn, F);
  fft_stage<2>(sre, sim, sgn, F);
}

// ===================== spectral distortion kernel (row-per-workgroup) =====================
__global__ void __launch_bounds__(BLK)
spectral_kernel(const float* __restrict__ in, float* __restrict__ out,
                int f0, int f1, int tm0, int tm1) {
  __shared__ float sre[ROWLEN];
  __shared__ float sim[ROWLEN];
  const size_t base = (size_t)blockIdx.x * ROWLEN;
  for (int i = threadIdx.x; i < ROWLEN; i += BLK) { sre[i] = in[base + i]; sim[i] = 0.0f; }
  __syncthreads();
  fft4096(sre, sim, -1.0f);                       // forward
  for (int i = threadIdx.x; i < ROWLEN; i += BLK)
    if (i >= f0 && i < f1) { sre[i] *= 0.1f; sim[i] *= 0.1f; }  // mag mask == complex scale
  __syncthreads();
  fft4096(sre, sim, 1.0f);                        // inverse (unscaled)
  const float inv = 1.0f / 4096.0f;
  for (int i = threadIdx.x; i < ROWLEN; i += BLK) {
    float v = sre[i] * inv;                       // .real
    if (i >= tm0 && i < tm1) v *= 0.1f;           // time mask
    out[base + i] = v;
  }
}

// ===================== wavelet compress kernel (db4, 4 levels) =====================
__global__ void __launch_bounds__(BLK)
wavelet_kernel(const float* __restrict__ in, float* __restrict__ out, float level) {
  __shared__ float bufA[ROWLEN];
  __shared__ float bufB[ROWLEN];
  __shared__ float bufD[3860];       // d1:2051 | d2:1029 | d3:518 | d4:262
  __shared__ float red[BLK];
  __shared__ float s_thr;
  const int tid = threadIdx.x;
  const size_t base = (size_t)blockIdx.x * ROWLEN;
  for (int i = tid; i < ROWLEN; i += BLK) bufA[i] = in[base + i];
  __syncthreads();

  const int dofs[4] = {0, 2051, 3080, 3598};
  const int dlen[4] = {2051, 1029, 518, 262};

  float* cur = bufA; float* nxt = bufB;
  int n = ROWLEN;
  // ---- analysis: 4 DWT levels, symmetric padding, stride 2 ----
  for (int lev = 0; lev < 4; ++lev) {
    const int m = (n + 7) >> 1;
    for (int j = tid; j < m; j += BLK) {
      float sa = 0.f, sd = 0.f;
#pragma unroll
      for (int s = 0; s < 8; ++s) {
        int t = 2 * j + 1 - s;
        int ts = (t < 0) ? (-t - 1) : ((t >= n) ? (2 * n - 1 - t) : t);
        float xv = cur[ts];
        sa += c_declo[s] * xv;
        sd += ((s & 1) ? c_declo[7 - s] : -c_declo[7 - s]) * xv;   // dec_hi
      }
      nxt[j] = sa;
      bufD[dofs[lev] + j] = sd;
    }
    __syncthreads();
    float* tmp = cur; cur = nxt; nxt = tmp;
    n = m;
  }
  // ---- threshold = level * std(a4), ddof=0 ----
  float loc = 0.f;
  for (int i = tid; i < n; i += BLK) loc += cur[i];
  red[tid] = loc; __syncthreads();
  for (int s2 = BLK / 2; s2; s2 >>= 1) { if (tid < s2) red[tid] += red[tid + s2]; __syncthreads(); }
  const float mean = red[0] / (float)n;
  __syncthreads();
  loc = 0.f;
  for (int i = tid; i < n; i += BLK) { float d = cur[i] - mean; loc += d * d; }
  red[tid] = loc; __syncthreads();
  for (int s2 = BLK / 2; s2; s2 >>= 1) { if (tid < s2) red[tid] += red[tid + s2]; __syncthreads(); }
  if (tid == 0) s_thr = level * sqrtf(red[0] / (float)n);
  __syncthreads();
  const float thr = s_thr;
  for (int i = tid; i < n; i += BLK) { float v = cur[i]; cur[i] = (fabsf(v) < thr) ? 0.f : v; }
  __syncthreads();
  // ---- synthesis: 4 IDWT levels (details hard-thresholded on read) ----
  for (int lev = 3; lev >= 0; --lev) {
    const int m = (n < dlen[lev]) ? n : dlen[lev];   // trim a to d length
    const int outlen = 2 * m - 6;
    const float* dd = bufD + dofs[lev];
    for (int j = tid; j < outlen; j += BLK) {
      float acc = 0.f;
#pragma unroll
      for (int s = 0; s < 8; ++s) {
        int t = j + 6 - s;
        if (t >= 0 && t <= 2 * m - 2 && !(t & 1)) {
          int hh = t >> 1;
          float av = cur[hh];
          float dv = dd[hh];
          dv = (fabsf(dv) < thr) ? 0.f : dv;
          acc += c_declo[7 - s] * av                           // rec_lo
               + ((s & 1) ? -c_declo[s] : c_declo[s]) * dv;    // rec_hi
        }
      }
      nxt[j] = acc;
    }
    __syncthreads();
    float* tmp = cur; cur = nxt; nxt = tmp;
    n = outlen;
  }
  for (int i = tid; i < ROWLEN; i += BLK) out[base + i] = cur[i];
}

// ===================== additive-noise kernel =====================
__device__ static inline float erfinv_f(float u) {   // Giles 2012 single precision
  float w = -logf((1.0f - u) * (1.0f + u));
  float p;
  if (w < 5.0f) {
    w -= 2.5f;
    p = 2.81022636e-08f;
    p = fmaf(p, w, 3.43273939e-07f);  p = fmaf(p, w, -3.5233877e-06f);
    p = fmaf(p, w, -4.39150654e-06f); p = fmaf(p, w, 0.00021858087f);
    p = fmaf(p, w, -0.00125372503f);  p = fmaf(p, w, -0.00417768164f);
    p = fmaf(p, w, 0.246640727f);     p = fmaf(p, w, 1.50140941f);
  } else {
    w = sqrtf(w) - 3.0f;
    p = -0.000200214257f;
    p = fmaf(p, w, 0.000100950558f);  p = fmaf(p, w, 0.00134934322f);
    p = fmaf(p, w, -0.00367342844f);  p = fmaf(p, w, 0.00573950773f);
    p = fmaf(p, w, -0.0076224613f);   p = fmaf(p, w, 0.00943887047f);
    p = fmaf(p, w, 1.00167406f);      p = fmaf(p, w, 1.00233146f);
  }
  return p * u;
}

__global__ void __launch_bounds__(BLK)
noise_kernel(const float* __restrict__ in, float* __restrict__ out,
             uint32_t k0, uint32_t k1, uint32_t halfN, float coef) {
  __shared__ float red[BLK];
  const int tid = threadIdx.x;
  const size_t base = (size_t)blockIdx.x * ROWLEN;
  // std with ddof=1 over the row
  float loc = 0.f;
  for (int i = tid; i < ROWLEN; i += BLK) loc += in[base + i];
  red[tid] = loc; __syncthreads();
  for (int s = BLK / 2; s; s >>= 1) { if (tid < s) red[tid] += red[tid + s]; __syncthreads(); }
  const float mean = red[0] * (1.0f / (float)ROWLEN);
  __syncthreads();
  loc = 0.f;
  for (int i = tid; i < ROWLEN; i += BLK) { float d = in[base + i] - mean; loc += d * d; }
  red[tid] = loc; __syncthreads();
  for (int s = BLK / 2; s; s >>= 1) { if (tid < s) red[tid] += red[tid + s]; __syncthreads(); }
  const float scale = coef * sqrtf(red[0] / (float)(ROWLEN - 1));
  __syncthreads();
  // JAX normal(key, shape): half-split counters, uniform->erfinv
  for (int i = tid; i < ROWLEN; i += BLK) {
    const uint32_t flat = (uint32_t)(base + i);
    uint32_t c0, c1, o0, o1, bits;
    if (flat < halfN) { c0 = flat; c1 = flat + halfN; tf2x32(k0, k1, c0, c1, &o0, &o1); bits = o0; }
    else              { c0 = flat - halfN; c1 = flat; tf2x32(k0, k1, c0, c1, &o0, &o1); bits = o1; }
    uint32_t fb = (bits >> 9) | 0x3f800000u;
    float f; __builtin_memcpy(&f, &fb, 4);
    float u01 = f - 1.0f;
    const float mn = -0.99999994f;                 // nextafter(-1, 0)
    float uu = fmaf(u01, 1.0f - mn, mn);
    uu = fmaxf(mn, uu);
    float nrm = 1.4142135623730951f * erfinv_f(uu);
    out[base + i] = in[base + i] + scale * nrm;
  }
}

// ===================== host launcher =====================
extern "C" void kernel_launch(void* const* d_in, const int* in_sizes, int n_in,
                              void* d_out, int out_size, void* d_ws, size_t ws_size,
                              hipStream_t stream) {
  (void)n_in; (void)out_size; (void)d_ws; (void)ws_size;
  const float* x = (const float*)d_in[0];
  float* o = (float*)d_out;
  const size_t NE = (size_t)in_sizes[0];          // 64*64*4096
  const int rows = (int)(NE / ROWLEN);            // 4096
  const uint32_t halfN = (uint32_t)(NE / 2);

  // keys: kd,kn,kcd,kcn = split(key(42), 4)  (key data = {0, 42})
  uint32_t sub[8];
  jax_split4(0u, 42u, sub);
  const uint32_t kd0 = sub[0], kd1 = sub[1], kn0 = sub[2], kn1 = sub[3];
  const uint32_t kcd0 = sub[4], kcd1 = sub[5], kcn0 = sub[6], kcn1 = sub[7];

  const int fw = 409;   // int(0.1*4096)
  const int tw = 204;   // int(0.05*4096)
  int f0[2], t0[2], f1[2], t1[2];
  const uint32_t kk[2][2] = {{kd0, kd1}, {kcd0, kcd1}};
  for (int q = 0; q < 2; ++q) {
    uint32_t s8[8];
    jax_split4(kk[q][0], kk[q][1], s8);           // k1,k2,k3,k4
    bool aF = bits_to_u01_host(jax_bits_scalar(s8[0], s8[1])) < 0.3f;
    int  vF = (int)(jax_bits_scalar(s8[2], s8[3]) % (uint32_t)(ROWLEN - fw));
    bool aT = bits_to_u01_host(jax_bits_scalar(s8[4], s8[5])) < 0.3f;
    int  vT = (int)(jax_bits_scalar(s8[6], s8[7]) % (uint32_t)(ROWLEN - tw));
    f0[q] = aF ? vF : 0;  f1[q] = aF ? vF + fw : 0;   // empty interval when not applied
    t0[q] = aT ? vT : 0;  t1[q] = aT ? vT + tw : 0;
  }

  // out0 = x
  hipMemcpyAsync(o, x, NE * sizeof(float), hipMemcpyDeviceToDevice, stream);
  // out1 = wavelet_compress(x, 0.5)
  wavelet_kernel<<<rows, BLK, 0, stream>>>(x, o + NE, 0.5f);
  // out2 = spectral_distortion(x, kd)
  spectral_kernel<<<rows, BLK, 0, stream>>>(x, o + 2 * NE, f0[0], f1[0], t0[0], t1[0]);
  // out3 = x + noise(kn)
  noise_kernel<<<rows, BLK, 0, stream>>>(x, o + 3 * NE, kn0, kn1, halfN, 0.02f);
  // out4 = add_noise(spectral(wavelet(x, 0.25), kcd), kcn)   (chained in place)
  wavelet_kernel<<<rows, BLK, 0, stream>>>(x, o + 4 * NE, 0.25f);
  spectral_kernel<<<rows, BLK, 0, stream>>>(o + 4 * NE, o + 4 * NE, f0[1], f1[1], t0[1], t1[1]);
  noise_kernel<<<rows, BLK, 0, stream>>>(o + 4 * NE, o + 4 * NE, kcn0, kcn1, halfN, 0.02f);
}